// FlaxLlamaAttention_80977313399058
// MI455X (gfx1250) — compile-verified
//
#include <hip/hip_runtime.h>
#include <hip/hip_bf16.h>

// ---------------------------------------------------------------------------
// FlaxLlamaAttention forward on MI455X (gfx1250, wave32, WMMA).
// fp32 reference -> fp32 tensor path: V_WMMA_F32_16X16X4_F32 everywhere.
// ---------------------------------------------------------------------------

typedef float v2f __attribute__((ext_vector_type(2)));
typedef float v8f __attribute__((ext_vector_type(8)));
typedef int i4v __attribute__((ext_vector_type(4)));

__device__ __forceinline__ v8f wmma_f32(v2f a, v2f b, v8f c) {
  // 8 args: (neg_a, A, neg_b, B, c_mod, C, reuse_a, reuse_b)
  return __builtin_amdgcn_wmma_f32_16x16x4_f32(false, a, false, b, (short)0, c,
                                               false, false);
}

// CDNA5 async global->LDS path (ASYNCcnt-tracked DMA-style copy).
// Probed signature: (AS1 v4i* src, AS3 v4i* dst, imm offset, imm cpol).
#if defined(__has_builtin)
#if __has_builtin(__builtin_amdgcn_global_load_async_to_lds_b128)
#define HAS_ASYNC_LDS 1
#endif
#endif
#ifndef HAS_ASYNC_LDS
#define HAS_ASYNC_LDS 0
#endif

#if defined(__has_builtin)
#if __has_builtin(__builtin_amdgcn_s_wait_asynccnt)
#define WAIT_ASYNC(n) __builtin_amdgcn_s_wait_asynccnt(n)
#endif
#endif
#ifndef WAIT_ASYNC
#define WAIT_ASYNC(n) asm volatile("s_wait_asynccnt " #n ::: "memory")
#endif

#if HAS_ASYNC_LDS
#define ASYNC_CP_B128(dst, src)                                      \
  __builtin_amdgcn_global_load_async_to_lds_b128(                    \
      (__attribute__((address_space(1))) i4v*)(src),                 \
      (__attribute__((address_space(3))) i4v*)(dst), 0, 0)
#endif

// CDNA5 16x16x4 f32 fragment layout (ISA 7.12.2):
//   A (16x4):  lanes 0-15 hold row M=lane, {K=0,K=1}; lanes 16-31 {K=2,K=3}.
//   B (4x16):  v[0] = {K=0 | K=2} across lanes, v[1] = {K=1 | K=3}.
//   C/D(16x16): vgpr r: lanes0-15 = (M=r, N=lane), lanes16-31 = (M=r+8).

// ===========================================================================
// GEMM: C[M,N] = A[M,K] @ B[K,N], fp32, WMMA 16x16x4.
// Block = 128 threads (4 waves). Block tile 128x64; wave tile 32x64
// (B-fragments reused across 2 M-subtiles). K-chunk 16, LDS double-buffered;
// staging via GLOBAL_LOAD_ASYNC_TO_LDS_B128 when available (else register
// prefetch), overlapping chunk c+1 traffic with chunk c WMMAs.
// EPI==0: plain row-major store. EPI==1: qkv de-interleave (col j -> j%3).
// ===========================================================================
template <int EPI>
__global__ __launch_bounds__(128) void gemm_f32_wmma(
    const float* __restrict__ A, const float* __restrict__ B,
    float* __restrict__ C0, float* __restrict__ C1, float* __restrict__ C2,
    int M, int N, int K) {
  constexpr int ASR = 20;  // A LDS row stride: 16B-aligned, conflict-free
  constexpr int BSR = 80;  // B LDS row stride: half-wave ranges disjoint mod 64
  constexpr int ABUF = 128 * ASR;
  constexpr int BBUF = 16 * BSR;
  __shared__ __align__(16) float As[2 * ABUF];  // 20 KB
  __shared__ __align__(16) float Bs[2 * BBUF];  // 10 KB

  const int t = threadIdx.x;
  const int w = t >> 5, lane = t & 31;
  const int hi = lane >> 4, l16 = lane & 15;
  const int m0 = blockIdx.y * 128, n0 = blockIdx.x * 64;

  // Staging coordinates: A tile 128x16 = 512 float4 (4/thread),
  //                      B tile 16x64  = 256 float4 (2/thread)
  const int arow = t >> 2, ac4 = t & 3;
  const int brow = t >> 4, bc4 = t & 15;

  v8f acc[2][4] = {};

  auto compute = [&](int buf) {
    const float* Ab = &As[buf * ABUF];
    const float* Bb = &Bs[buf * BBUF];
#pragma unroll
    for (int ks = 0; ks < 16; ks += 4) {
      v2f a0 = *(const v2f*)&Ab[(w * 32 + l16) * ASR + ks + 2 * hi];
      v2f a1 = *(const v2f*)&Ab[(w * 32 + 16 + l16) * ASR + ks + 2 * hi];
#pragma unroll
      for (int nt = 0; nt < 4; ++nt) {
        v2f b;
        b.x = Bb[(ks + 0 + 2 * hi) * BSR + nt * 16 + l16];
        b.y = Bb[(ks + 1 + 2 * hi) * BSR + nt * 16 + l16];
        acc[0][nt] = wmma_f32(a0, b, acc[0][nt]);
        acc[1][nt] = wmma_f32(a1, b, acc[1][nt]);
      }
    }
  };

  const int nch = K >> 4;

#if HAS_ASYNC_LDS
  auto stage_async = [&](int buf, int k0) {
#pragma unroll
    for (int r = 0; r < 4; ++r) {
      int row = r * 32 + arow;
      ASYNC_CP_B128(&As[buf * ABUF + row * ASR + ac4 * 4],
                    A + (size_t)(m0 + row) * K + k0 + ac4 * 4);
    }
#pragma unroll
    for (int r = 0; r < 2; ++r) {
      int row = r * 8 + brow;
      ASYNC_CP_B128(&Bs[buf * BBUF + row * BSR + bc4 * 4],
                    B + (size_t)(k0 + row) * N + n0 + bc4 * 4);
    }
  };
  stage_async(0, 0);
  for (int c = 0; c < nch; ++c) {
    if (c + 1 < nch) {
      stage_async((c + 1) & 1, (c + 1) << 4);
      WAIT_ASYNC(6);  // retire chunk c's 6 async ops; keep chunk c+1 in flight
    } else {
      WAIT_ASYNC(0);
    }
    __syncthreads();
    compute(c & 1);
    __syncthreads();
  }
#else
  float4 ra[4], rb[2];
  auto load_regs = [&](int k0) {
#pragma unroll
    for (int r = 0; r < 4; ++r)
      ra[r] = *(const float4*)(A + (size_t)(m0 + r * 32 + arow) * K + k0 +
                               ac4 * 4);
#pragma unroll
    for (int r = 0; r < 2; ++r)
      rb[r] = *(const float4*)(B + (size_t)(k0 + r * 8 + brow) * N + n0 +
                               bc4 * 4);
  };
  load_regs(0);
  for (int c = 0; c < nch; ++c) {
    const int buf = c & 1;
#pragma unroll
    for (int r = 0; r < 4; ++r)
      *(float4*)&As[buf * ABUF + (r * 32 + arow) * ASR + ac4 * 4] = ra[r];
#pragma unroll
    for (int r = 0; r < 2; ++r)
      *(float4*)&Bs[buf * BBUF + (r * 8 + brow) * BSR + bc4 * 4] = rb[r];
    __syncthreads();
    if (c + 1 < nch) load_regs((c + 1) << 4);  // overlaps with compute below
    compute(buf);
    __syncthreads();
  }
#endif

#pragma unroll
  for (int mt = 0; mt < 2; ++mt)
#pragma unroll
    for (int nt = 0; nt < 4; ++nt)
#pragma unroll
      for (int rr = 0; rr < 8; ++rr) {
        int row = m0 + w * 32 + mt * 16 + rr + 8 * hi;
        int col = n0 + nt * 16 + l16;
        float v = acc[mt][nt][rr];
        if (EPI == 0) {
          C0[(size_t)row * N + col] = v;
        } else {
          int mm = col % 3, ii = col / 3;  // qkv[..., d, 3] de-interleave
          float* dst = (mm == 0) ? C0 : (mm == 1) ? C1 : C2;
          dst[(size_t)row * 4096 + ii] = v;
        }
      }
}

// ===========================================================================
// RoPE in place on q and k, layout [S, H*128]. Pair (p, p+64) per thread.
// ===========================================================================
__global__ __launch_bounds__(256) void rope_kernel(float* __restrict__ q,
                                                   float* __restrict__ k,
                                                   const int* __restrict__ pos) {
  int idx = blockIdx.x * 256 + threadIdx.x;  // S*H*64 total
  int s = idx >> 11;
  int rem = idx & 2047;
  int head = rem >> 6, p = rem & 63;
  float tt = (float)pos[s];
  float th = tt * __expf(-(float)p * 0.14391156463f);  // ln(1e4)/64
  float c = __cosf(th), sn = __sinf(th);
  size_t off = (size_t)s * 4096 + head * 128;
  float x1 = q[off + p], x2 = q[off + p + 64];
  q[off + p] = x1 * c - x2 * sn;
  q[off + p + 64] = x2 * c + x1 * sn;
  x1 = k[off + p];
  x2 = k[off + p + 64];
  k[off + p] = x1 * c - x2 * sn;
  k[off + p + 64] = x2 * c + x1 * sn;
}

// ===========================================================================
// Flash attention, one block = (64 q rows) x (1 head), 4 waves x 16 q rows.
// St = K_blk @ Q^T (A-fragment of Q == B-fragment of Q^T, so Q stays in
// registers). Ct = V^T @ Pt accumulated transposed. Per-lane softmax stats
// (lane%16 == q column), cross-half combine via shfl_xor(16).
// ===========================================================================
__global__ __launch_bounds__(128) void attn_kernel(
    const float* __restrict__ Q, const float* __restrict__ Kb,
    const float* __restrict__ Vb, float* __restrict__ O) {
  constexpr int KS = 132;  // K LDS row stride (kseq x 128 feat)
  constexpr int VS = 36;   // V^T LDS row stride (feat x 32 kseq)
  __shared__ __align__(16) float smem[32 * KS + 128 * VS];  // 35,328 B
  float* Ks = smem;
  float* Vt = smem + 32 * KS;

  const int t = threadIdx.x, w = t >> 5, lane = t & 31;
  const int hi = lane >> 4, l16 = lane & 15;
  const int s0 = blockIdx.x * 64, head = blockIdx.y;
  const float scale = 0.08838834764831845f;  // 1/sqrt(128)

  v2f qf[32];
  {
    const float* Qp =
        Q + (size_t)(s0 + w * 16 + l16) * 4096 + head * 128 + 2 * hi;
#pragma unroll
    for (int f = 0; f < 32; ++f) {
      v2f v = *(const v2f*)(Qp + 4 * f);
      qf[f].x = v.x * scale;
      qf[f].y = v.y * scale;
    }
  }

  v8f acc[8] = {};  // Ct[d][q]: row d = dt*16+rr+8*hi, col q = l16
  float mrow = -1e30f, lrow = 0.f;
  const int qg = s0 + w * 16 + l16;
  const int nkb = (s0 + 64) >> 5;

  for (int kb = 0; kb < nkb; ++kb) {
    const int k0 = kb * 32;
    __syncthreads();
#pragma unroll
    for (int r = 0; r < 8; ++r) {
      int id = r * 128 + t;
      int row = id >> 5, c4 = id & 31;
      float4 kv = *(const float4*)(Kb + (size_t)(k0 + row) * 4096 +
                                   head * 128 + c4 * 4);
      *(float4*)&Ks[row * KS + c4 * 4] = kv;
      float4 vv = *(const float4*)(Vb + (size_t)(k0 + row) * 4096 +
                                   head * 128 + c4 * 4);
      Vt[(c4 * 4 + 0) * VS + row] = vv.x;
      Vt[(c4 * 4 + 1) * VS + row] = vv.y;
      Vt[(c4 * 4 + 2) * VS + row] = vv.z;
      Vt[(c4 * 4 + 3) * VS + row] = vv.w;
    }
    __syncthreads();

    v8f st[2];
#pragma unroll
    for (int kt = 0; kt < 2; ++kt) {
      v8f c = {};
#pragma unroll
      for (int f = 0; f < 32; ++f) {
        v2f a = *(const v2f*)&Ks[(kt * 16 + l16) * KS + 4 * f + 2 * hi];
        c = wmma_f32(a, qf[f], c);
      }
      st[kt] = c;
    }

    float mloc = -1e30f;
#pragma unroll
    for (int kt = 0; kt < 2; ++kt)
#pragma unroll
      for (int rr = 0; rr < 8; ++rr) {
        int kgl = k0 + kt * 16 + rr + 8 * hi;
        float sv = (kgl <= qg) ? st[kt][rr] : -1e30f;
        st[kt][rr] = sv;
        mloc = fmaxf(mloc, sv);
      }
    mloc = fmaxf(mloc, __shfl_xor(mloc, 16));
    float mnew = fmaxf(mrow, mloc);
    float corr = __expf(mrow - mnew);
    mrow = mnew;

    float ssum = 0.f;
#pragma unroll
    for (int kt = 0; kt < 2; ++kt)
#pragma unroll
      for (int rr = 0; rr < 8; ++rr) {
        float p = __expf(st[kt][rr] - mnew);
        st[kt][rr] = p;
        ssum += p;
      }
    ssum += __shfl_xor(ssum, 16);
    lrow = lrow * corr + ssum;
#pragma unroll
    for (int i = 0; i < 8; ++i)
#pragma unroll
      for (int rr = 0; rr < 8; ++rr) acc[i][rr] *= corr;

    // Ct += V^T @ Pt; B-frag g covers Pt rows 4g..4g+3 (kseq in [0,32)).
#pragma unroll
    for (int g = 0; g < 8; ++g) {
      const int kt = g >> 2, fp = g & 3;
      const int base = (fp & 1) * 4;
      v2f b;
      if (fp < 2) {  // rows 4fp..4fp+3 held by lanes 0-15
        float ox = __shfl_xor(st[kt][base + 2], 16);
        float oy = __shfl_xor(st[kt][base + 3], 16);
        b.x = hi ? ox : st[kt][base + 0];
        b.y = hi ? oy : st[kt][base + 1];
      } else {  // rows 8+base.. held by lanes 16-31
        float ox = __shfl_xor(st[kt][base + 0], 16);
        float oy = __shfl_xor(st[kt][base + 1], 16);
        b.x = hi ? st[kt][base + 2] : ox;
        b.y = hi ? st[kt][base + 3] : oy;
      }
#pragma unroll
      for (int dt = 0; dt < 8; ++dt) {
        v2f a = *(const v2f*)&Vt[(dt * 16 + l16) * VS + 4 * g + 2 * hi];
        acc[dt] = wmma_f32(a, b, acc[dt]);
      }
    }
  }

  __syncthreads();
  float rl = 1.0f / lrow;
  float* Tw = smem + w * (16 * KS);
#pragma unroll
  for (int dt = 0; dt < 8; ++dt)
#pragma unroll
    for (int rr = 0; rr < 8; ++rr)
      Tw[l16 * KS + dt * 16 + rr + 8 * hi] = acc[dt][rr] * rl;
  __syncthreads();
#pragma unroll
  for (int r = 0; r < 16; ++r) {
    float4 v = *(const float4*)&Tw[r * KS + lane * 4];
    *(float4*)(O + (size_t)(s0 + w * 16 + r) * 4096 + head * 128 + lane * 4) =
        v;
  }
}

// ===========================================================================
extern "C" void kernel_launch(void* const* d_in, const int* in_sizes, int n_in,
                              void* d_out, int out_size, void* d_ws,
                              size_t ws_size, hipStream_t stream) {
  (void)in_sizes; (void)n_in; (void)out_size; (void)ws_size;
  const float* hidden = (const float*)d_in[0];
  // d_in[1] = attention_mask (causal tril) -- known analytically, unused
  const int* pos = (const int*)d_in[2];
  const float* Wqkv = (const float*)d_in[3];
  const float* Wo = (const float*)d_in[4];
  float* out = (float*)d_out;

  const size_t SD = (size_t)2048 * 4096;
  float* qb = (float*)d_ws;  // [S, H*128]
  float* kb = qb + SD;
  float* vb = kb + SD;
  float* ctx = vb + SD;  // total 128 MB of d_ws

  // 1) fused QKV projection with de-interleave epilogue
  gemm_f32_wmma<1><<<dim3(12288 / 64, 2048 / 128), 128, 0, stream>>>(
      hidden, Wqkv, qb, kb, vb, 2048, 12288, 4096);
  // 2) RoPE on q, k in place
  rope_kernel<<<dim3((2048 * 32 * 64) / 256), 256, 0, stream>>>(qb, kb, pos);
  // 3) causal flash attention -> ctx
  attn_kernel<<<dim3(2048 / 64, 32), 128, 0, stream>>>(qb, kb, vb, ctx);
  // 4) output projection -> d_out
  gemm_f32_wmma<0><<<dim3(4096 / 64, 2048 / 128), 128, 0, stream>>>(
      ctx, Wo, out, nullptr, nullptr, 2048, 4096, 4096);
}